// GNNLinkPredictor_70763881169321
// MI455X (gfx1250) — compile-verified
//
#include <hip/hip_runtime.h>

typedef __attribute__((ext_vector_type(2))) float v2f;
typedef __attribute__((ext_vector_type(8))) float v8f;

// ---------------------------------------------------------------------------
// Utility: zero a float buffer
// ---------------------------------------------------------------------------
__global__ void zero_f32(float* __restrict__ p, long n) {
  long i = (long)blockIdx.x * blockDim.x + threadIdx.x;
  if (i < n) p[i] = 0.0f;
}

// ---------------------------------------------------------------------------
// h = emb[x]   (N rows of 64 floats, float4-vectorized)
// ---------------------------------------------------------------------------
__global__ void gather_rows(const float* __restrict__ emb, const int* __restrict__ x,
                            float* __restrict__ h, int N) {
  int i = blockIdx.x * blockDim.x + threadIdx.x;  // one per float4, N*16 total
  if (i >= N * 16) return;
  int node = i >> 4;
  int c = (i & 15) << 2;
  float4 v = *(const float4*)(emb + (long)x[node] * 64 + c);
  *(float4*)(h + (long)node * 64 + c) = v;
}

// ---------------------------------------------------------------------------
// deg[dst[e]] += 1.0
// ---------------------------------------------------------------------------
__global__ void degree_kernel(const int* __restrict__ dst, float* __restrict__ deg, int E) {
  int e = blockIdx.x * blockDim.x + threadIdx.x;
  if (e < E) atomicAdd(&deg[dst[e]], 1.0f);
}

// ---------------------------------------------------------------------------
// agg[dst[e], :] += h[src[e], :]   one thread per (edge, 4-float chunk)
// ---------------------------------------------------------------------------
template <int Din>
__global__ void scatter_add(const float* __restrict__ h, const int* __restrict__ src,
                            const int* __restrict__ dst, float* __restrict__ agg, int E) {
  const int chunks = Din / 4;
  long i = (long)blockIdx.x * blockDim.x + threadIdx.x;
  int e = (int)(i / chunks);
  int c = (int)(i % chunks) * 4;
  if (e >= E) return;
  int s = src[e];
  int d = dst[e];
  float4 v = *(const float4*)(h + (long)s * Din + c);
  float* p = agg + (long)d * Din + c;
  atomicAdd(p + 0, v.x);
  atomicAdd(p + 1, v.y);
  atomicAdd(p + 2, v.z);
  atomicAdd(p + 3, v.w);
}

// ---------------------------------------------------------------------------
// out = act( (agg/deg) @ Wl + h @ Wr + bias )
// One wave computes one 16x16 output tile with V_WMMA_F32_16X16X4_F32.
// blockDim.x = 128 (4 waves); grid = (N/16, Dout/64).
//
// f32 WMMA operand layouts (ISA 7.12.2):
//   A 16x4 : lane M = lane%16; VGPR pair holds K = 2*(lane/16), 2*(lane/16)+1
//   B 4x16 : lane N = lane%16; VGPR pair holds K = 2*(lane/16), 2*(lane/16)+1
//   C/D    : VGPR r holds row M = r + 8*(lane/16), col N = lane%16
// ---------------------------------------------------------------------------
template <int Din, int Dout, bool RELU>
__global__ __launch_bounds__(128) void sage_combine(
    const float* __restrict__ agg, const float* __restrict__ hin,
    const float* __restrict__ deg, const float* __restrict__ Wl,
    const float* __restrict__ Wr, const float* __restrict__ bias,
    float* __restrict__ out) {
  const int lane = threadIdx.x & 31;
  const int wave = threadIdx.x >> 5;
  const int tm = blockIdx.x;                 // node tile (16 rows)
  const int tn = blockIdx.y * 4 + wave;      // column tile (16 cols)
  const int mn = lane & 15;                  // A-row (M) and B/C-column (N)
  const int kg = lane >> 4;                  // K-pair group: 0 or 1

  const long row = (long)tm * 16 + mn;
  const int col = tn * 16 + mn;
  const float invdeg = 1.0f / fmaxf(deg[row], 1.0f);

  const float* __restrict__ arow = agg + row * Din;
  const float* __restrict__ hrow = hin + row * Din;

  v8f acc = {};

  // (agg * invdeg) @ Wl
#pragma unroll
  for (int k = 0; k < Din; k += 4) {
    const int k0 = k + 2 * kg;
    float2 t = *(const float2*)(arow + k0);
    v2f a;
    a.x = t.x * invdeg;
    a.y = t.y * invdeg;
    v2f b;
    b.x = Wl[(long)k0 * Dout + col];
    b.y = Wl[(long)(k0 + 1) * Dout + col];
    acc = __builtin_amdgcn_wmma_f32_16x16x4_f32(false, a, false, b, (short)0,
                                                acc, false, false);
  }

  // h @ Wr
#pragma unroll
  for (int k = 0; k < Din; k += 4) {
    const int k0 = k + 2 * kg;
    float2 t = *(const float2*)(hrow + k0);
    v2f a;
    a.x = t.x;
    a.y = t.y;
    v2f b;
    b.x = Wr[(long)k0 * Dout + col];
    b.y = Wr[(long)(k0 + 1) * Dout + col];
    acc = __builtin_amdgcn_wmma_f32_16x16x4_f32(false, a, false, b, (short)0,
                                                acc, false, false);
  }

  const float bv = bias[col];
#pragma unroll
  for (int r = 0; r < 8; ++r) {
    float v = acc[r] + bv;
    if (RELU) v = fmaxf(v, 0.0f);
    out[((long)tm * 16 + r + 8 * kg) * Dout + col] = v;
  }
}

// ---------------------------------------------------------------------------
// out[l] = dot(z[eli0[l]], z[eli1[l]])  over 64 dims
// ---------------------------------------------------------------------------
__global__ void decode_kernel(const float* __restrict__ z, const int* __restrict__ eli,
                              float* __restrict__ out, int L) {
  int l = blockIdx.x * blockDim.x + threadIdx.x;
  if (l >= L) return;
  int a = eli[l];
  int b = eli[L + l];
  const float4* pa = (const float4*)(z + (long)a * 64);
  const float4* pb = (const float4*)(z + (long)b * 64);
  float s = 0.0f;
#pragma unroll
  for (int k = 0; k < 16; ++k) {
    float4 u = pa[k];
    float4 v = pb[k];
    s += u.x * v.x + u.y * v.y + u.z * v.z + u.w * v.w;
  }
  out[l] = s;
}

// ---------------------------------------------------------------------------
// Launch sequence
// ---------------------------------------------------------------------------
extern "C" void kernel_launch(void* const* d_in, const int* in_sizes, int n_in,
                              void* d_out, int out_size, void* d_ws, size_t ws_size,
                              hipStream_t stream) {
  const int* x    = (const int*)d_in[0];
  const int* ei   = (const int*)d_in[1];
  const int* eli  = (const int*)d_in[2];
  const float* emb = (const float*)d_in[3];
  const float* W1l = (const float*)d_in[4];
  const float* b1  = (const float*)d_in[5];
  const float* W1r = (const float*)d_in[6];
  const float* W2l = (const float*)d_in[7];
  const float* b2  = (const float*)d_in[8];
  const float* W2r = (const float*)d_in[9];

  const int N = in_sizes[0];
  const int E = in_sizes[1] / 2;
  const int L = in_sizes[2] / 2;
  const int* src = ei;
  const int* dst = ei + E;

  // Workspace layout (floats): h[N*64] | h1[N*128] | agg[N*128] | deg[N]
  float* ws  = (float*)d_ws;
  float* h   = ws;
  float* h1  = ws + (long)N * 64;
  float* agg = ws + (long)N * 192;
  float* deg = ws + (long)N * 320;
  float* z   = h;  // reuse: h is dead after layer-1 combine

  const int B = 256;

  // h = emb[x]
  gather_rows<<<(N * 16 + B - 1) / B, B, 0, stream>>>(emb, x, h, N);

  // degree (shared by both layers)
  zero_f32<<<(N + B - 1) / B, B, 0, stream>>>(deg, N);
  degree_kernel<<<(E + B - 1) / B, B, 0, stream>>>(dst, deg, E);

  // ---- Layer 1: agg = segsum(h[src] by dst); h1 = relu(mean@W1l + b1 + h@W1r)
  zero_f32<<<((long)N * 64 + B - 1) / B, B, 0, stream>>>(agg, (long)N * 64);
  {
    long nt = (long)E * 16;
    scatter_add<64><<<(nt + B - 1) / B, B, 0, stream>>>(h, src, dst, agg, E);
  }
  {
    dim3 g(N / 16, 2);  // 128 output cols / (4 waves * 16)
    sage_combine<64, 128, true><<<g, 128, 0, stream>>>(agg, h, deg, W1l, W1r, b1, h1);
  }

  // ---- Layer 2: agg = segsum(h1[src] by dst); z = mean@W2l + b2 + h1@W2r
  zero_f32<<<((long)N * 128 + B - 1) / B, B, 0, stream>>>(agg, (long)N * 128);
  {
    long nt = (long)E * 32;
    scatter_add<128><<<(nt + B - 1) / B, B, 0, stream>>>(h1, src, dst, agg, E);
  }
  {
    dim3 g(N / 16, 1);
    sage_combine<128, 64, false><<<g, 128, 0, stream>>>(agg, h1, deg, W2l, W2r, b2, z);
  }

  // ---- Decode
  decode_kernel<<<(L + B - 1) / B, B, 0, stream>>>(z, eli, (float*)d_out, L);
}